// KVCache_652835029298
// MI455X (gfx1250) — compile-verified
//
#include <hip/hip_runtime.h>
#include <stdint.h>

#define H      512
#define CSIZE  65536
#define NLEV   3
#define NCHUNK 64
#define ROWS_PER_CHUNK (CSIZE / NCHUNK)   // 1024

typedef __attribute__((ext_vector_type(2))) float v2f;
typedef __attribute__((ext_vector_type(8))) float v8f;

// ---------- workspace layout (in floats unless noted) ----------
// [0]        imp_new_mean (f32)
// [1]        min key (u32, monotonic-encoded float)
// [2]        max key (u32)
// [4..6]     counts[3] (u32)
// [8..519]   v[512]  (mean-pooled value_in)
// [520 ..)   partials: NCHUNK x 3 x 512 f32
// after that levels: CSIZE bytes (u8)
#define WS_PART_BASE 520
#define WS_PART_FLTS (NCHUNK * NLEV * H)

__device__ __forceinline__ unsigned f2key(float f) {
    unsigned u = __float_as_uint(f);
    return (u & 0x80000000u) ? ~u : (u | 0x80000000u);
}
__device__ __forceinline__ float key2f(unsigned k) {
    unsigned b = (k & 0x80000000u) ? (k & 0x7FFFFFFFu) : ~k;
    return __uint_as_float(b);
}
__device__ __forceinline__ int level_of(float imp, float mn, float mx) {
    float impn = (imp - mn) / (mx - mn + 1e-8f);
    float lv = rintf((1.0f - impn) * (float)(NLEV - 1));  // round-half-even, like jnp.round
    int l = (int)lv;
    return l < 0 ? 0 : (l > NLEV - 1 ? NLEV - 1 : l);
}

// K0: init workspace scalars
__global__ void init_ws_k(uint32_t* wsu) {
    if (threadIdx.x == 0) {
        wsu[1] = 0xFFFFFFFFu;  // min key accumulator
        wsu[2] = 0u;           // max key accumulator
        wsu[4] = 0u; wsu[5] = 0u; wsu[6] = 0u;  // counts
    }
}

// K1: v = mean(value_in over B*S rows); ws[0] = mean(importance_new)
__global__ __launch_bounds__(256) void pool_k(const float* __restrict__ value_in,
                                              const float* __restrict__ imp_new,
                                              float* __restrict__ wsf) {
    int h = blockIdx.x * 256 + threadIdx.x;   // grid 2 blocks -> h in [0,512)
    float acc = 0.0f;
    for (int n = 0; n < 1024; ++n) acc += value_in[n * H + h];
    wsf[8 + h] = acc * (1.0f / 1024.0f);
    if (blockIdx.x == 0) {
        __shared__ float s[256];
        s[threadIdx.x] = imp_new[threadIdx.x];
        __syncthreads();
        for (int st = 128; st > 0; st >>= 1) {
            if (threadIdx.x < st) s[threadIdx.x] += s[threadIdx.x + st];
            __syncthreads();
        }
        if (threadIdx.x == 0) wsf[0] = s[0] * (1.0f / 256.0f);
    }
}

// K2: min/max of cache_importance with row idx replaced by ws[0]
__global__ __launch_bounds__(256) void minmax_k(const float* __restrict__ ci,
                                                const float* __restrict__ wsf,
                                                uint32_t* __restrict__ wsu,
                                                const int* __restrict__ idxp) {
    __shared__ unsigned smin[256], smax[256];
    int idx = *idxp;
    float repl = wsf[0];
    unsigned kmin = 0xFFFFFFFFu, kmax = 0u;
    int stride = gridDim.x * blockDim.x;
    for (int n = blockIdx.x * blockDim.x + threadIdx.x; n < CSIZE; n += stride) {
        float f = ci[n];
        if (n == idx) f = repl;
        unsigned k = f2key(f);
        kmin = min(kmin, k);
        kmax = max(kmax, k);
    }
    smin[threadIdx.x] = kmin; smax[threadIdx.x] = kmax;
    __syncthreads();
    for (int st = 128; st > 0; st >>= 1) {
        if (threadIdx.x < st) {
            smin[threadIdx.x] = min(smin[threadIdx.x], smin[threadIdx.x + st]);
            smax[threadIdx.x] = max(smax[threadIdx.x], smax[threadIdx.x + st]);
        }
        __syncthreads();
    }
    if (threadIdx.x == 0) {
        atomicMin(&wsu[1], smin[0]);
        atomicMax(&wsu[2], smax[0]);
    }
}

// K3: per-row level -> u8 array, and per-level counts (integer atomics: exact)
__global__ __launch_bounds__(256) void levels_k(const float* __restrict__ ci,
                                                const float* __restrict__ wsf,
                                                uint32_t* __restrict__ wsu,
                                                unsigned char* __restrict__ levels,
                                                const int* __restrict__ idxp) {
    __shared__ unsigned lc[NLEV];
    if (threadIdx.x < NLEV) lc[threadIdx.x] = 0u;
    __syncthreads();
    int idx = *idxp;
    float repl = wsf[0];
    float mn = key2f(wsu[1]);
    float mx = key2f(wsu[2]);
    int stride = gridDim.x * blockDim.x;
    for (int n = blockIdx.x * blockDim.x + threadIdx.x; n < CSIZE; n += stride) {
        float f = ci[n];
        if (n == idx) f = repl;
        int l = level_of(f, mn, mx);
        levels[n] = (unsigned char)l;
        atomicAdd(&lc[l], 1u);
    }
    __syncthreads();
    if (threadIdx.x < NLEV) atomicAdd(&wsu[4 + threadIdx.x], lc[threadIdx.x]);
}

// K4: per-level masked column sums via V_WMMA_F32_16X16X4_F32.
//   s[l,h] = sum_n onehot[n,l] * cv[n,h]      (row-idx fixup applied in finalize_k)
//   A[m,k] = (level[k]==m), B[k,j] = cv[k, colbase+j], C accum 16x16.
// One wave per 16-column tile per row chunk; 16 rows / 4 accumulators per
// iteration: unconditional clause-able loads, no branches, WMMA RAW broken.
__global__ __launch_bounds__(256) void levelsum_wmma_k(const float* __restrict__ cv,
                                                       const unsigned char* __restrict__ levels,
                                                       float* __restrict__ partials) {
    int lane = threadIdx.x & 31;
    int wave = threadIdx.x >> 5;                 // 0..7
    int coltile = blockIdx.x * 8 + wave;         // 0..31
    int colbase = coltile * 16;
    int chunk = blockIdx.y;                      // 0..NCHUNK-1
    int row0 = chunk * ROWS_PER_CHUNK;

    int m  = lane & 15;                          // A: M index / B: column index
    int kb = (lane >> 4) * 2;                    // K sub-index per ISA A/B 16x4 layout
    int col = colbase + m;

    v8f c0 = {}, c1 = {}, c2 = {}, c3 = {};
    const float* base = cv + (size_t)(row0 + kb) * H + col;
    const unsigned char* lv = levels + row0 + kb;

    for (int it = 0; it < ROWS_PER_CHUNK / 16; ++it) {
        // level bytes for the 8 K-rows this lane contributes to (4 groups of 4 rows)
        unsigned short p0 = *(const unsigned short*)(lv + 0);
        unsigned short p1 = *(const unsigned short*)(lv + 4);
        unsigned short p2 = *(const unsigned short*)(lv + 8);
        unsigned short p3 = *(const unsigned short*)(lv + 12);
        // cv tile: 8 unconditional dword loads (rows kb, kb+1 of each 4-row group)
        float x0 = base[0 * H];
        float x1 = base[1 * H];
        float x2 = base[4 * H];
        float x3 = base[5 * H];
        float x4 = base[8 * H];
        float x5 = base[9 * H];
        float x6 = base[12 * H];
        float x7 = base[13 * H];

        v2f a0, a1, a2, a3, b0, b1, b2, b3;
        a0.x = ((p0 & 0xff) == m) ? 1.0f : 0.0f;
        a0.y = ((p0 >> 8)   == m) ? 1.0f : 0.0f;
        a1.x = ((p1 & 0xff) == m) ? 1.0f : 0.0f;
        a1.y = ((p1 >> 8)   == m) ? 1.0f : 0.0f;
        a2.x = ((p2 & 0xff) == m) ? 1.0f : 0.0f;
        a2.y = ((p2 >> 8)   == m) ? 1.0f : 0.0f;
        a3.x = ((p3 & 0xff) == m) ? 1.0f : 0.0f;
        a3.y = ((p3 >> 8)   == m) ? 1.0f : 0.0f;
        b0.x = x0; b0.y = x1;
        b1.x = x2; b1.y = x3;
        b2.x = x4; b2.y = x5;
        b3.x = x6; b3.y = x7;

        c0 = __builtin_amdgcn_wmma_f32_16x16x4_f32(false, a0, false, b0, (short)0, c0, false, false);
        c1 = __builtin_amdgcn_wmma_f32_16x16x4_f32(false, a1, false, b1, (short)0, c1, false, false);
        c2 = __builtin_amdgcn_wmma_f32_16x16x4_f32(false, a2, false, b2, (short)0, c2, false, false);
        c3 = __builtin_amdgcn_wmma_f32_16x16x4_f32(false, a3, false, b3, (short)0, c3, false, false);

        base += 16 * H;
        lv   += 16;
    }
    v8f c = (c0 + c1) + (c2 + c3);
    // C layout: VGPR r, lanes 0-15 hold row M=r -> rows 0..2 are the level sums
    if (lane < 16) {
        partials[((size_t)chunk * NLEV + 0) * H + colbase + lane] = c[0];
        partials[((size_t)chunk * NLEV + 1) * H + colbase + lane] = c[1];
        partials[((size_t)chunk * NLEV + 2) * H + colbase + lane] = c[2];
    }
}

// K5: reduce chunk partials (fixed order), apply the scattered-row fixup
// (sum used original cv; replace row idx's contribution with v), run the 3
// affine pyramid paths on the per-level sums (biases scaled by counts),
// add LoRA residual, divide by N.
__global__ __launch_bounds__(512) void finalize_k(const float* __restrict__ partials,
                                                  const uint32_t* __restrict__ wsu,
                                                  const float* __restrict__ wsf,
                                                  const float* __restrict__ cv,
                                                  const unsigned char* __restrict__ levels,
                                                  const int* __restrict__ idxp,
                                                  const float* __restrict__ Wc0, const float* __restrict__ bc0,
                                                  const float* __restrict__ Wd0, const float* __restrict__ bd0,
                                                  const float* __restrict__ Wc1, const float* __restrict__ bc1,
                                                  const float* __restrict__ Wd1, const float* __restrict__ bd1,
                                                  const float* __restrict__ Wc2, const float* __restrict__ bc2,
                                                  const float* __restrict__ Wd2, const float* __restrict__ bd2,
                                                  const float* __restrict__ loraA, const float* __restrict__ loraB,
                                                  float* __restrict__ out) {
    __shared__ float sL[NLEV][H];
    __shared__ float bufA[H], bufB[H];
    __shared__ float tot[H];
    __shared__ float tA[4];
    int tid = threadIdx.x;

    for (int e = tid; e < NLEV * H; e += 512) {
        float acc = 0.0f;
        for (int ch = 0; ch < NCHUNK; ++ch) acc += partials[(size_t)ch * (NLEV * H) + e];
        sL[e / H][e % H] = acc;
    }
    // Scattered-row fixup: thread tid owns column tid for all three levels,
    // so this touches only values written by this same thread (no race).
    {
        int idx = *idxp;
        int lvl = (int)levels[idx];
        sL[lvl][tid] += wsf[8 + tid] - cv[(size_t)idx * H + tid];
    }
    tot[tid] = 0.0f;
    __syncthreads();

    const float* Wc[NLEV]  = {Wc0, Wc1, Wc2};
    const float* bcv[NLEV] = {bc0, bc1, bc2};
    const float* Wd[NLEV]  = {Wd0, Wd1, Wd2};
    const float* bdv[NLEV] = {bd0, bd1, bd2};
    const int dims[4] = {512, 256, 128, 64};

    for (int l = 0; l < NLEV; ++l) {
        float cnt = (float)wsu[4 + l];
        bufA[tid] = sL[l][tid];
        __syncthreads();
        float* cur = bufA; float* nxt = bufB;
        for (int j = 0; j <= l; ++j) {             // compress: Wc[j] is [dims[j], dims[j+1]]
            int fi = dims[j], fo = dims[j + 1];
            if (tid < fo) {
                float acc = cnt * bcv[j][tid];
                for (int i = 0; i < fi; ++i) acc += cur[i] * Wc[j][i * fo + tid];
                nxt[tid] = acc;
            }
            __syncthreads();
            float* t = cur; cur = nxt; nxt = t;
        }
        for (int j = l; j >= 0; --j) {             // decompress: Wd[j] is [dims[j+1], dims[j]]
            int fi = dims[j + 1], fo = dims[j];
            if (tid < fo) {
                float acc = cnt * bdv[j][tid];
                for (int i = 0; i < fi; ++i) acc += cur[i] * Wd[j][i * fo + tid];
                nxt[tid] = acc;
            }
            __syncthreads();
            float* t = cur; cur = nxt; nxt = t;
        }
        tot[tid] += cur[tid];
        __syncthreads();
    }

    if (tid < 4) {
        float acc = 0.0f;
        for (int h2 = 0; h2 < H; ++h2) acc += tot[h2] * loraA[h2 * 4 + tid];
        tA[tid] = acc;
    }
    __syncthreads();
    float lora = tA[0] * loraB[0 * H + tid] + tA[1] * loraB[1 * H + tid] +
                 tA[2] * loraB[2 * H + tid] + tA[3] * loraB[3 * H + tid];
    out[tid] = (tot[tid] + lora * 0.25f) * (1.0f / (float)CSIZE);
}

extern "C" void kernel_launch(void* const* d_in, const int* in_sizes, int n_in,
                              void* d_out, int out_size, void* d_ws, size_t ws_size,
                              hipStream_t stream) {
    const float* value_in = (const float*)d_in[1];
    const float* imp_new  = (const float*)d_in[2];
    const float* cv       = (const float*)d_in[4];
    const float* ci       = (const float*)d_in[5];
    const float* Wc0 = (const float*)d_in[6],  *bc0 = (const float*)d_in[7];
    const float* Wd0 = (const float*)d_in[8],  *bd0 = (const float*)d_in[9];
    const float* Wc1 = (const float*)d_in[10], *bc1 = (const float*)d_in[11];
    const float* Wd1 = (const float*)d_in[12], *bd1 = (const float*)d_in[13];
    const float* Wc2 = (const float*)d_in[14], *bc2 = (const float*)d_in[15];
    const float* Wd2 = (const float*)d_in[16], *bd2 = (const float*)d_in[17];
    const float* loraA = (const float*)d_in[18];
    const float* loraB = (const float*)d_in[19];
    const int*   idxp  = (const int*)d_in[20];

    float*    wsf = (float*)d_ws;
    uint32_t* wsu = (uint32_t*)d_ws;
    float*    partials = wsf + WS_PART_BASE;
    unsigned char* levels = (unsigned char*)(wsf + WS_PART_BASE + WS_PART_FLTS);

    init_ws_k<<<1, 64, 0, stream>>>(wsu);
    pool_k<<<2, 256, 0, stream>>>(value_in, imp_new, wsf);
    minmax_k<<<64, 256, 0, stream>>>(ci, wsf, wsu, idxp);
    levels_k<<<64, 256, 0, stream>>>(ci, wsf, wsu, levels, idxp);
    levelsum_wmma_k<<<dim3(4, NCHUNK), 256, 0, stream>>>(cv, levels, partials);
    finalize_k<<<1, 512, 0, stream>>>(partials, wsu, wsf, cv, levels, idxp,
                                      Wc0, bc0, Wd0, bd0,
                                      Wc1, bc1, Wd1, bd1,
                                      Wc2, bc2, Wd2, bd2,
                                      loraA, loraB, (float*)d_out);
}